// PhaseBlock_6983616823512
// MI455X (gfx1250) — compile-verified
//
#include <hip/hip_runtime.h>
#include <math.h>

#define B_      4096
#define D_      1024
#define K_      4096
#define ACTIVE_ 409

typedef float v2f __attribute__((ext_vector_type(2)));
typedef float v8f __attribute__((ext_vector_type(8)));

// ---------------------------------------------------------------------------
// Kernel 1: inv_norm[k] = 1 / max(||P_k||_2, eps)   (complex row norm over D)
// ---------------------------------------------------------------------------
__global__ __launch_bounds__(256) void pnorm_kernel(const float* __restrict__ Pr,
                                                    const float* __restrict__ Pi,
                                                    float* __restrict__ inv_norm) {
    __shared__ float red[256];
    const int k = blockIdx.x;
    const int t = threadIdx.x;
    const float4 a = *(const float4*)&Pr[k * D_ + t * 4];
    const float4 b = *(const float4*)&Pi[k * D_ + t * 4];
    red[t] = a.x * a.x + a.y * a.y + a.z * a.z + a.w * a.w +
             b.x * b.x + b.y * b.y + b.z * b.z + b.w * b.w;
    __syncthreads();
    for (int off = 128; off > 0; off >>= 1) {
        if (t < off) red[t] += red[t + off];
        __syncthreads();
    }
    if (t == 0) inv_norm[k] = 1.0f / fmaxf(sqrtf(red[0]), 1e-12f);
}

// ---------------------------------------------------------------------------
// Kernel 2: scores[b,k] = (zr·Pr_k + zi·Pi_k) * inv_norm[k]
// 128x128 tile / block, 8 waves, FP32 WMMA 16x16x4 (exact — keeps top-k order)
// ---------------------------------------------------------------------------
__global__ __launch_bounds__(256) void scores_kernel(const float* __restrict__ zr,
                                                     const float* __restrict__ zi,
                                                     const float* __restrict__ Pr,
                                                     const float* __restrict__ Pi,
                                                     const float* __restrict__ inv_norm,
                                                     float* __restrict__ scores) {
    __shared__ float As[128 * 36];   // z tile  (m x 32), stride 36: 16B-aligned rows, no bank conflicts
    __shared__ float Bs[128 * 36];   // P tile  (n x 32)
    const int m0 = blockIdx.y * 128;
    const int n0 = blockIdx.x * 128;
    const int tid  = threadIdx.x;
    const int lane = tid & 31;
    const int wave = tid >> 5;
    const int wm = wave & 3;     // 4 waves along M (32 rows each)
    const int wn = wave >> 2;    // 2 waves along N (64 cols each)
    const int half = lane >> 4;  // A/B fragment half-wave K offset
    const int l    = lane & 15;

    v8f acc[2][4] = {};

    for (int ch = 0; ch < 64; ++ch) {
        const float* Zp = (ch < 32) ? zr : zi;
        const float* Pp = (ch < 32) ? Pr : Pi;
        const int dbase = (ch & 31) * 32;
        #pragma unroll
        for (int i = 0; i < 4; ++i) {
            const int f   = tid + 256 * i;
            const int row = f >> 3;
            const int c4  = (f & 7) << 2;
            *(float4*)&As[row * 36 + c4] = *(const float4*)&Zp[(m0 + row) * D_ + dbase + c4];
            *(float4*)&Bs[row * 36 + c4] = *(const float4*)&Pp[(n0 + row) * D_ + dbase + c4];
        }
        __syncthreads();
        #pragma unroll
        for (int kk = 0; kk < 32; kk += 4) {
            // ISA A layout (f32 16x4): lanes 0-15 -> K = kk..kk+1, lanes 16-31 -> K = kk+2..kk+3
            const int dA = kk + half * 2;
            v2f a[2], bb[4];
            #pragma unroll
            for (int im = 0; im < 2; ++im)
                a[im] = *(const v2f*)&As[(wm * 32 + im * 16 + l) * 36 + dA];
            #pragma unroll
            for (int in = 0; in < 4; ++in)
                bb[in] = *(const v2f*)&Bs[(wn * 64 + in * 16 + l) * 36 + dA];
            #pragma unroll
            for (int im = 0; im < 2; ++im)
                #pragma unroll
                for (int in = 0; in < 4; ++in)
                    acc[im][in] = __builtin_amdgcn_wmma_f32_16x16x4_f32(
                        false, a[im], false, bb[in], (short)0, acc[im][in], false, false);
        }
        __syncthreads();
    }

    // C layout: VGPR r -> (M = r + 8*half, N = lane%16); scale by inv_norm and store
    #pragma unroll
    for (int in = 0; in < 4; ++in) {
        const int n = n0 + wn * 64 + in * 16 + l;
        const float sc = inv_norm[n];
        #pragma unroll
        for (int im = 0; im < 2; ++im) {
            const int mBase = m0 + wm * 32 + im * 16 + half * 8;
            #pragma unroll
            for (int r = 0; r < 8; ++r)
                scores[(mBase + r) * K_ + n] = acc[im][in][r] * sc;
        }
    }
}

// ---------------------------------------------------------------------------
// Kernel 3: per row — radix-select top-409 threshold, phase scatter, residual,
//           row renorm to sqrt(K), write (B,K,2)
// ---------------------------------------------------------------------------
__device__ __forceinline__ unsigned fkey(float x) {
    unsigned u = __float_as_uint(x);
    return u ^ ((u >> 31) ? 0xFFFFFFFFu : 0x80000000u);  // monotone order-preserving
}

__global__ __launch_bounds__(256) void topk_out_kernel(const float* __restrict__ scores,
                                                       const float* __restrict__ zr,
                                                       const float* __restrict__ zi,
                                                       const float* __restrict__ phi,
                                                       const float* __restrict__ lam_r,
                                                       const float* __restrict__ lam_i,
                                                       float* __restrict__ out) {
    __shared__ float sval[K_];
    __shared__ float souti[K_];
    __shared__ unsigned hist[256];
    __shared__ unsigned scan[256];
    __shared__ float red[256];
    __shared__ unsigned sh_prefix, sh_T;

    const int b = blockIdx.x;
    const int t = threadIdx.x;

    for (int i = 0; i < 16; ++i) {
        const int k = t + 256 * i;                 // coalesced
        sval[k] = scores[b * K_ + k];
    }
    if (t == 0) { sh_prefix = 0u; sh_T = ACTIVE_; }
    __syncthreads();

    // MSB-first 8-bit radix select of the ACTIVE-th largest key
    for (int pass = 0; pass < 4; ++pass) {
        const unsigned prefix = sh_prefix;
        const unsigned T      = sh_T;
        const int shift       = 24 - 8 * pass;
        const unsigned hmask  = (pass == 0) ? 0u : (0xFFFFFFFFu << (shift + 8));
        hist[t] = 0;
        __syncthreads();
        for (int i = 0; i < 16; ++i) {
            const unsigned kk = fkey(sval[t * 16 + i]);
            if ((kk & hmask) == (prefix & hmask))
                atomicAdd(&hist[(kk >> shift) & 0xFFu], 1u);
        }
        __syncthreads();
        if (t == 0) {
            unsigned cum = 0;
            for (int d = 255; d >= 0; --d) {
                const unsigned c = hist[d];
                if (cum + c >= T) { sh_prefix = prefix | ((unsigned)d << shift); sh_T = T - cum; break; }
                cum += c;
            }
        }
        __syncthreads();
    }
    const unsigned thresh = sh_prefix;   // key of the ACTIVE-th largest value
    const unsigned Tfin   = sh_T;        // how many exact-threshold elems to keep

    // deterministic index-ordered ranking among keys equal to the threshold
    unsigned eq = 0;
    for (int i = 0; i < 16; ++i)
        eq += (fkey(sval[t * 16 + i]) == thresh) ? 1u : 0u;
    scan[t] = eq;
    __syncthreads();
    for (int off = 1; off < 256; off <<= 1) {
        const unsigned v   = scan[t];
        const unsigned add = (t >= off) ? scan[t - off] : 0u;
        __syncthreads();
        scan[t] = v + add;
        __syncthreads();
    }
    unsigned rank = (t == 0) ? 0u : scan[t - 1];

    const float lr = lam_r[0], li = lam_i[0];
    float local = 0.0f;
    for (int i = 0; i < 16; ++i) {
        const int k = t * 16 + i;
        const float v = sval[k];
        const unsigned kk = fkey(v);
        bool act;
        if (kk > thresh)        act = true;
        else if (kk == thresh) { act = (rank < Tfin); rank++; }
        else                    act = false;

        float s, c;
        __sincosf(phi[k], &s, &c);
        float ar = act ? v * c : 0.0f;
        float ai = act ? v * s : 0.0f;
        if (k < D_) {                                    // residual: lam * z (zero-padded to K)
            const float a  = zr[b * D_ + k];
            const float bz = zi[b * D_ + k];
            ar += lr * a - li * bz;
            ai += lr * bz + li * a;
        }
        sval[k]  = ar;
        souti[k] = ai;
        local += ar * ar + ai * ai;
    }
    red[t] = local;
    __syncthreads();
    for (int off = 128; off > 0; off >>= 1) {
        if (t < off) red[t] += red[t + off];
        __syncthreads();
    }
    const float scale = 64.0f / fmaxf(sqrtf(red[0]), 1e-12f);  // sqrt(K)=64
    float2* o = (float2*)out;
    for (int i = 0; i < 16; ++i) {
        const int k = t + 256 * i;                 // coalesced float2 stores
        o[b * K_ + k] = make_float2(sval[k] * scale, souti[k] * scale);
    }
}

// ---------------------------------------------------------------------------
extern "C" void kernel_launch(void* const* d_in, const int* in_sizes, int n_in,
                              void* d_out, int out_size, void* d_ws, size_t ws_size,
                              hipStream_t stream) {
    const float* zr   = (const float*)d_in[0];
    const float* zi   = (const float*)d_in[1];
    const float* Pr   = (const float*)d_in[2];
    const float* Pi   = (const float*)d_in[3];
    const float* phi  = (const float*)d_in[4];
    const float* lamr = (const float*)d_in[5];
    const float* lami = (const float*)d_in[6];
    float* out = (float*)d_out;

    float* ws       = (float*)d_ws;
    float* inv_norm = ws;            // K_ floats
    float* scores   = ws + K_;       // B_*K_ floats (~67 MB)

    pnorm_kernel<<<K_, 256, 0, stream>>>(Pr, Pi, inv_norm);
    dim3 grid(K_ / 128, B_ / 128);
    scores_kernel<<<grid, 256, 0, stream>>>(zr, zi, Pr, Pi, inv_norm, scores);
    topk_out_kernel<<<B_, 256, 0, stream>>>(scores, zr, zi, phi, lamr, lami, out);
}